// GraphSAGE_45208825757772
// MI455X (gfx1250) — compile-verified
//
#include <hip/hip_runtime.h>

typedef float v2f __attribute__((ext_vector_type(2)));
typedef float v8f __attribute__((ext_vector_type(8)));

// ---------------------------------------------------------------------------
// zero fill
// ---------------------------------------------------------------------------
__global__ void k_zero(float* __restrict__ p, long long n) {
  long long i = (long long)blockIdx.x * blockDim.x + threadIdx.x;
  if (i < n) p[i] = 0.0f;
}

// ---------------------------------------------------------------------------
// in-degree: deg[dst] += 1
// ---------------------------------------------------------------------------
__global__ void k_degree(const long long* __restrict__ ei, long long E,
                         float* __restrict__ deg) {
  long long e = (long long)blockIdx.x * blockDim.x + threadIdx.x;
  if (e < E) atomicAdd(&deg[(int)ei[E + e]], 1.0f);
}

// ---------------------------------------------------------------------------
// C[M x ncols] = A[M x K] @ concat(B0[K x nh], B1[K x nh])   (row-major, fp32)
// One wave computes a 16-row x 128-col strip with V_WMMA_F32_16X16X4_F32
// (8 column tiles, 8 f32 accumulators = 64 VGPRs -> no spills, good
// occupancy). B is staged in LDS in KC=64-row chunks, stored as adjacent-K
// pairs per column so each lane's {B[k][c], B[k+1][c]} fragment is one
// 8-byte ds read (pairs combine into ds_load_2addr_b64). Per K-step:
// 1 global b64 (A) + 4 DS ops + 8 WMMA.
// grid.x covers row tiles (4 waves/block), grid.y covers 128-col groups.
// Requires M % 16 == 0, K % 64 == 0 (true here: M=50000, K=512/64).
// ---------------------------------------------------------------------------
#define GEMM_KC 64
#define GEMM_NT 8   // 8 col tiles = 128 cols per wave

__global__ void k_gemm_wmma(const float* __restrict__ A, int lda,
                            const float* __restrict__ B0,
                            const float* __restrict__ B1, int nh,
                            float* __restrict__ C, int ldc, int M, int K) {
  __shared__ float ldsB[GEMM_KC * 128];  // 32 KB, paired-K layout

  const int tid  = threadIdx.x;          // 0..127
  const int lane = tid & 31;
  const int wave = tid >> 5;
  const int rowTile = blockIdx.x * 4 + wave;          // wave-uniform
  const bool active = (rowTile * 16) < M;             // wave-uniform
  const int colBase = blockIdx.y * 128;

  const int mrow  = rowTile * 16 + (lane & 15);
  const int khalf = (lane >> 4) << 1;    // 0 (lanes 0-15) or 2 (lanes 16-31)
  const int ncol  = lane & 15;

  v8f acc[GEMM_NT];
#pragma unroll
  for (int t = 0; t < GEMM_NT; ++t) acc[t] = (v8f){};

  for (int kbase = 0; kbase < K; kbase += GEMM_KC) {
    // ---- cooperative stage of B[kbase:kbase+KC, colBase:colBase+128] ----
    // iteration ii loads B row (kbase+ii), column (colBase+tid): coalesced.
    {
      const int gcol = colBase + tid;
      const float* __restrict__ Bsel = (gcol < nh) ? B0 : B1;
      const int bcol = (gcol < nh) ? gcol : (gcol - nh);
#pragma unroll 4
      for (int ii = 0; ii < GEMM_KC; ++ii) {
        float v = Bsel[(size_t)(kbase + ii) * nh + bcol];
        ldsB[((ii >> 1) * 128 + tid) * 2 + (ii & 1)] = v;
      }
    }
    __syncthreads();

    if (active) {
#pragma unroll 4
      for (int kk = 0; kk < GEMM_KC; kk += 4) {
        // A fragment: row mrow, cols kbase+kk+khalf, +1 (one b64 load)
        v2f a = *(const v2f*)&A[(size_t)mrow * lda + kbase + kk + khalf];
        const int kp = ((kk + khalf) >> 1) * 128;  // paired-row base in LDS
#pragma unroll
        for (int t = 0; t < GEMM_NT; ++t) {
          v2f b = *(const v2f*)&ldsB[(kp + t * 16 + ncol) * 2];
          acc[t] = __builtin_amdgcn_wmma_f32_16x16x4_f32(
              false, a, false, b, (short)0, acc[t], false, false);
        }
      }
    }
    __syncthreads();
  }

  if (active) {
    // C/D layout: VGPR r -> lanes 0-15: M=r, lanes 16-31: M=r+8; N = lane&15
    const int rbase = rowTile * 16 + ((lane >> 4) << 3);
#pragma unroll
    for (int r = 0; r < 8; ++r) {
      float* crow = &C[(size_t)(rbase + r) * ldc + colBase + ncol];
#pragma unroll
      for (int t = 0; t < GEMM_NT; ++t) crow[t * 16] = acc[t][r];
    }
  }
}

// ---------------------------------------------------------------------------
// agg[dst, 0:nf] += feat[src, 0:nf]  per edge; one thread = 4 floats.
// cshift = log2(nf/4); total threads = E << cshift.
// ---------------------------------------------------------------------------
__global__ void k_scatter(const long long* __restrict__ ei, long long E,
                          const float* __restrict__ feat, int fstride,
                          float* __restrict__ agg, int nf, int cshift) {
  long long gid = (long long)blockIdx.x * blockDim.x + threadIdx.x;
  long long e = gid >> cshift;
  if (e >= E) return;
  int c4 = (int)(gid & ((1 << cshift) - 1)) * 4;
  int s = (int)ei[e];
  int d = (int)ei[E + e];
  float4 v = *(const float4*)&feat[(size_t)s * fstride + c4];
  float* p = &agg[(size_t)d * nf + c4];
  atomicAdd(p + 0, v.x);
  atomicAdd(p + 1, v.y);
  atomicAdd(p + 2, v.z);
  atomicAdd(p + 3, v.w);
}

// ---------------------------------------------------------------------------
// h = relu(agg1/max(deg,1) + b1 + xr)   (in-place over agg1; xr = y1 cols 64..127)
// ---------------------------------------------------------------------------
__global__ void k_combine1(const float* __restrict__ y1, const float* __restrict__ deg,
                           const float* __restrict__ b1, float* __restrict__ h,
                           long long N) {
  long long gid = (long long)blockIdx.x * blockDim.x + threadIdx.x;
  if (gid >= N * 64) return;
  long long i = gid >> 6;
  int f = (int)(gid & 63);
  float d = fmaxf(deg[i], 1.0f);
  float v = h[gid] / d + b1[f] + y1[(size_t)i * 128 + 64 + f];
  h[gid] = fmaxf(v, 0.0f);
}

// ---------------------------------------------------------------------------
// out2 = agg2/max(deg,1) + b2 + hr  (hr = y2 cols 128..255), fused with
// graph mean-pool accumulation: pooled[batch[i]] += out2; cnt[batch[i]] += 1
// ---------------------------------------------------------------------------
__global__ void k_pool(const float* __restrict__ agg2, const float* __restrict__ y2,
                       const float* __restrict__ deg, const float* __restrict__ b2,
                       const long long* __restrict__ batch,
                       float* __restrict__ pooled, float* __restrict__ cnt,
                       long long N) {
  long long gid = (long long)blockIdx.x * blockDim.x + threadIdx.x;
  if (gid >= N * 128) return;
  long long i = gid >> 7;
  int f = (int)(gid & 127);
  float d = fmaxf(deg[i], 1.0f);
  float v = agg2[gid] / d + b2[f] + y2[(size_t)i * 256 + 128 + f];
  int g = (int)batch[i];
  atomicAdd(&pooled[g * 128 + f], v);
  if (f == 0) atomicAdd(&cnt[g], 1.0f);
}

// ---------------------------------------------------------------------------
// logits = (pooled/cnt) @ w_fc + b_fc; log_softmax. 64 graphs -> 64 threads.
// ---------------------------------------------------------------------------
__global__ void k_head(const float* __restrict__ pooled, const float* __restrict__ cnt,
                       const float* __restrict__ wfc, const float* __restrict__ bfc,
                       float* __restrict__ out) {
  int g = threadIdx.x;  // blockDim.x == 64
  float c = fmaxf(cnt[g], 1.0f);
  float logit[10];
#pragma unroll
  for (int k = 0; k < 10; ++k) logit[k] = bfc[k];
  for (int f = 0; f < 128; ++f) {
    float p = pooled[g * 128 + f] / c;
#pragma unroll
    for (int k = 0; k < 10; ++k) logit[k] += p * wfc[f * 10 + k];
  }
  float m = logit[0];
#pragma unroll
  for (int k = 1; k < 10; ++k) m = fmaxf(m, logit[k]);
  float s = 0.0f;
#pragma unroll
  for (int k = 0; k < 10; ++k) s += expf(logit[k] - m);
  float lse = logf(s) + m;
#pragma unroll
  for (int k = 0; k < 10; ++k) out[g * 10 + k] = logit[k] - lse;
}

// ---------------------------------------------------------------------------
extern "C" void kernel_launch(void* const* d_in, const int* in_sizes, int n_in,
                              void* d_out, int out_size, void* d_ws, size_t ws_size,
                              hipStream_t stream) {
  (void)n_in; (void)out_size; (void)ws_size;
  const float*     x     = (const float*)d_in[0];
  const long long* ei    = (const long long*)d_in[1];  // int64 [2, E]
  const long long* batch = (const long long*)d_in[2];  // int64 [N]
  const float* w1l = (const float*)d_in[3];
  const float* b1l = (const float*)d_in[4];
  const float* w1r = (const float*)d_in[5];
  const float* w2l = (const float*)d_in[6];
  const float* b2l = (const float*)d_in[7];
  const float* w2r = (const float*)d_in[8];
  const float* wfc = (const float*)d_in[9];
  const float* bfc = (const float*)d_in[10];
  float* out = (float*)d_out;

  const long long N = in_sizes[0] / 512;   // 50000
  const long long E = in_sizes[1] / 2;     // 800000
  const long long G = 64;

  // workspace layout (floats): ~115.4 MB total
  float* ws     = (float*)d_ws;
  float* deg    = ws;                  // N
  float* y1     = deg    + N;          // N*128  (cols 0-63: x@w1_l, 64-127: x@w1_r)
  float* agg1   = y1     + N * 128;    // N*64   (reused as h after combine1)
  float* y2     = agg1   + N * 64;     // N*256  (cols 0-127: h@w2_l, 128-255: h@w2_r)
  float* agg2   = y2     + N * 256;    // N*128
  float* pooled = agg2   + N * 128;    // G*128
  float* cnt    = pooled + G * 128;    // G

  auto cdiv = [](long long a, long long b) { return (unsigned)((a + b - 1) / b); };

  // zero the accumulation buffers
  k_zero<<<cdiv(N, 256), 256, 0, stream>>>(deg, N);
  k_zero<<<cdiv(N * 64, 256), 256, 0, stream>>>(agg1, N * 64);
  k_zero<<<cdiv(N * 128, 256), 256, 0, stream>>>(agg2, N * 128);
  k_zero<<<cdiv(G * 129, 256), 256, 0, stream>>>(pooled, G * 128 + G);

  k_degree<<<cdiv(E, 256), 256, 0, stream>>>(ei, E, deg);

  const int rowTiles = (int)((N + 15) / 16);          // 3125

  // layer 1: y1 = x @ [w1_l | w1_r]  (M=N, K=512, 128 cols -> 1 col group;
  // x streamed from HBM exactly once)
  {
    dim3 grid(cdiv(rowTiles, 4), 1);
    k_gemm_wmma<<<grid, 128, 0, stream>>>(x, 512, w1l, w1r, 64, y1, 128, (int)N, 512);
  }
  // agg1[dst] += y1[src, 0:64]
  k_scatter<<<cdiv(E * 16, 256), 256, 0, stream>>>(ei, E, y1, 128, agg1, 64, 4);
  // h = relu(agg1/deg + b1 + y1[:,64:128])  in place
  k_combine1<<<cdiv(N * 64, 256), 256, 0, stream>>>(y1, deg, b1l, agg1, N);

  // layer 2: y2 = h @ [w2_l | w2_r]  (M=N, K=64, 256 cols -> 2 col groups;
  // h is L2-resident so the second pass over A is cheap)
  {
    dim3 grid(cdiv(rowTiles, 4), 2);
    k_gemm_wmma<<<grid, 128, 0, stream>>>(agg1, 64, w2l, w2r, 128, y2, 256, (int)N, 64);
  }
  // agg2[dst] += y2[src, 0:128]
  k_scatter<<<cdiv(E * 32, 256), 256, 0, stream>>>(ei, E, y2, 256, agg2, 128, 5);

  // fused layer-2 combine + graph mean-pool accumulation
  k_pool<<<cdiv(N * 128, 256), 256, 0, stream>>>(agg2, y2, deg, b2l, batch, pooled, cnt, N);

  // classifier head + log_softmax
  k_head<<<1, 64, 0, stream>>>(pooled, cnt, wfc, bfc, out);
}